// GraphCNN_89507118449100
// MI455X (gfx1250) — compile-verified
//
#include <hip/hip_runtime.h>

#define NN   4096
#define HID  128
#define NG   8
#define NLAY 2
#define KC   64          // j-chunk staged in LDS
#define HS_STRIDE 68     // 64 + 4 pad: keeps 16B alignment, staggers banks

static constexpr float LRELU_A = 0.01f;
static constexpr float BN_EPS  = 1e-5f;

typedef __attribute__((ext_vector_type(2))) float v2f;
typedef __attribute__((ext_vector_type(8))) float v8f;
typedef int vs4i __attribute__((vector_size(16)));   // <4 x i32> for async b128

#ifndef __has_builtin
#define __has_builtin(x) 0
#endif
#if __has_builtin(__builtin_amdgcn_global_load_async_to_lds_b128) && \
    __has_builtin(__builtin_amdgcn_s_wait_asynccnt)
#define USE_ASYNC 1
#else
#define USE_ASYNC 0
#endif

// 16B global -> LDS copy; async (ASYNCcnt-tracked DMA) when available.
__device__ __forceinline__ void cp16(const float* g, float* l) {
#if USE_ASYNC
    __builtin_amdgcn_global_load_async_to_lds_b128(
        (__attribute__((address_space(1))) vs4i*)(void*)g,
        (__attribute__((address_space(3))) vs4i*)(void*)l, 0, 0);
#else
    *(float4*)l = *(const float4*)g;
#endif
}
__device__ __forceinline__ void async_wait_keep9() {
#if USE_ASYNC
    __builtin_amdgcn_s_wait_asynccnt(9);   // retire previous chunk (9 ops/wave)
#endif
}
__device__ __forceinline__ void async_wait_all() {
#if USE_ASYNC
    __builtin_amdgcn_s_wait_asynccnt(0);
#endif
}

// ---------------------------------------------------------------------------
__global__ void prep_att_kernel(const float* __restrict__ Watt,
                                const float* __restrict__ aatt,
                                float* __restrict__ wa1, float* __restrict__ wa2) {
    int k = threadIdx.x;                    // 128 threads
    float v1 = 0.f, v2 = 0.f;
    for (int m = 0; m < 36; ++m) {
        float w = Watt[k * 36 + m];
        v1 += w * aatt[m];
        v2 += w * aatt[36 + m];
    }
    wa1[k] = v1;
    wa2[k] = v2;
}

__global__ void scores_kernel(const float* __restrict__ h,
                              const float* __restrict__ wa1,
                              const float* __restrict__ wa2,
                              float* __restrict__ s1, float* __restrict__ s2) {
    int wave = (blockIdx.x * blockDim.x + threadIdx.x) >> 5;
    int l    = threadIdx.x & 31;
    if (wave >= NN) return;
    const float* hr = h + (size_t)wave * HID;
    float a1 = 0.f, a2 = 0.f;
    for (int k = l; k < HID; k += 32) {
        float hv = hr[k];
        a1 += hv * wa1[k];
        a2 += hv * wa2[k];
    }
    for (int off = 16; off; off >>= 1) {
        a1 += __shfl_xor(a1, off, 32);
        a2 += __shfl_xor(a2, off, 32);
    }
    if (l == 0) { s1[wave] = a1; s2[wave] = a2; }
}

__global__ void max_kernel(const float* __restrict__ s2, float* __restrict__ out) {
    __shared__ float sm[256];
    float m = -3.4e38f;
    for (int i = threadIdx.x; i < NN; i += 256) m = fmaxf(m, s2[i]);
    sm[threadIdx.x] = m;
    __syncthreads();
    for (int off = 128; off; off >>= 1) {
        if (threadIdx.x < off) sm[threadIdx.x] = fmaxf(sm[threadIdx.x], sm[threadIdx.x + off]);
        __syncthreads();
    }
    if (threadIdx.x == 0) out[0] = sm[0];
}

// ---------------------------------------------------------------------------
// pooled = (adj ⊙ softmax(e)) @ h + (1+eps)*h.
// 4 waves/block: 64 rows x 64 cols. h[j:j+64, n0:n0+64] double-buffered in LDS
// via async global->LDS DMA; softmax denominator fused into the K loop.
__global__ __launch_bounds__(128) void agg_wmma_kernel(
        const float* __restrict__ adj, const float* __restrict__ h,
        const float* __restrict__ s1, const float* __restrict__ s2,
        const float* __restrict__ maxs2, const float* __restrict__ eps,
        int layer, float* __restrict__ out) {
    __shared__ alignas(16) float hs[2][KC * HS_STRIDE];
    __shared__ alignas(16) float s2s[2][KC];

    const int tid   = threadIdx.x;
    const int l     = tid & 31;
    const int rt    = l & 15;       // row-in-tile (A) / col-in-tile (B,C)
    const int khalf = l >> 4;       // K-pair selector
    const int i0    = blockIdx.x * 64 + (tid >> 5) * 16;
    const int n0    = blockIdx.y * 64;
    const int row   = i0 + rt;

    const float M   = maxs2[0];
    const float s1r = s1[row];
    float tt = s1r + M;
    const float rowmax = (tt > 0.f) ? tt : LRELU_A * tt;

    v8f acc[4] = {};
    float dsum = 0.f;
    const float* adjrow = adj + (size_t)row * NN;

    // stage h[jc:jc+KC, n0:n0+64] and s2[jc:jc+KC]; exactly 9 copy instr/wave
    auto stage = [&](int jc, int buf) {
#pragma unroll
        for (int t = 0; t < 8; ++t) {
            int idx = tid + t * 128;          // 0..1023 float4s
            int r = idx >> 4, c4 = idx & 15;
            cp16(h + (size_t)(jc + r) * HID + n0 + c4 * 4,
                 &hs[buf][r * HS_STRIDE + c4 * 4]);
        }
        if ((tid & 7) == 0) {                 // 4 lanes/wave -> 1 instr/wave
            int q = tid >> 3;                 // 0..15
            cp16(s2 + jc + q * 4, &s2s[buf][q * 4]);
        }
    };

    auto compute = [&](int jc, int buf) {
        const float* hsb = hs[buf];
        const float* s2b = s2s[buf];
#pragma unroll 4
        for (int j2 = 0; j2 < KC; j2 += 4) {
            const int k0 = j2 + khalf * 2;
            float e0 = s1r + s2b[k0];
            float e1 = s1r + s2b[k0 + 1];
            e0 = (e0 > 0.f) ? e0 : LRELU_A * e0;
            e1 = (e1 > 0.f) ? e1 : LRELU_A * e1;
            float w0 = __expf(e0 - rowmax);
            float w1 = __expf(e1 - rowmax);
            dsum += w0 + w1;                          // unmasked denom
            v2f a;
            a.x = adjrow[jc + k0]     * w0;
            a.y = adjrow[jc + k0 + 1] * w1;
#pragma unroll
            for (int t4 = 0; t4 < 4; ++t4) {
                v2f b;
                b.x = hsb[k0 * HS_STRIDE       + t4 * 16 + rt];
                b.y = hsb[(k0 + 1) * HS_STRIDE + t4 * 16 + rt];
                acc[t4] = __builtin_amdgcn_wmma_f32_16x16x4_f32(
                    false, a, false, b, (short)0, acc[t4], false, false);
            }
        }
    };

    constexpr int NCH = NN / KC;
    stage(0, 0);
    for (int k = 0; k < NCH; ++k) {
        const int buf = k & 1;
        if (k + 1 < NCH) { stage((k + 1) * KC, buf ^ 1); async_wait_keep9(); }
        else             { async_wait_all(); }
        __syncthreads();                 // chunk k resident for all waves
        compute(k * KC, buf);
        __syncthreads();                 // done reading buf before it's refilled
    }

    dsum += __shfl_xor(dsum, 16, 32);    // combine the two K-halves per row
    const float ep = 1.f + eps[layer];
#pragma unroll
    for (int t4 = 0; t4 < 4; ++t4) {
        const int col = n0 + t4 * 16 + rt;
#pragma unroll
        for (int v = 0; v < 8; ++v) {
            const int r = v + 8 * khalf;
            const float D = __shfl(dsum, r, 32);
            out[(size_t)(i0 + r) * HID + col] =
                acc[t4][v] / D + ep * h[(size_t)(i0 + r) * HID + col];
        }
    }
}

// ---------------------------------------------------------------------------
// C[M,128] = A'[M,128] @ W[128,128] + bias ; MODE 1: A' = relu(A*scale+shift)
template <int MODE>
__global__ __launch_bounds__(32) void gemm_wmma_kernel(
        const float* __restrict__ A, const float* __restrict__ W,
        const float* __restrict__ bias,
        const float* __restrict__ scale, const float* __restrict__ shift,
        float* __restrict__ C) {
    const int l     = threadIdx.x;
    const int rt    = l & 15;
    const int khalf = l >> 4;
    const int i0    = blockIdx.x * 16;
    const int n0    = blockIdx.y * 16;
    const int row   = i0 + rt;
    const int col   = n0 + rt;

    v8f acc = {};
    for (int k = 0; k < HID; k += 4) {
        const int k0 = k + khalf * 2;
        v2f a, b;
        a.x = A[(size_t)row * HID + k0];
        a.y = A[(size_t)row * HID + k0 + 1];
        if (MODE == 1) {
            float x0 = a.x * scale[k0] + shift[k0];
            float x1 = a.y * scale[k0 + 1] + shift[k0 + 1];
            a.x = (x0 > 0.f) ? x0 : 0.f;
            a.y = (x1 > 0.f) ? x1 : 0.f;
        }
        b.x = W[(size_t)k0 * HID + col];
        b.y = W[(size_t)(k0 + 1) * HID + col];
        acc = __builtin_amdgcn_wmma_f32_16x16x4_f32(
            false, a, false, b, (short)0, acc, false, false);
    }
    const float bs = bias[col];
#pragma unroll
    for (int v = 0; v < 8; ++v) {
        const int r = v + 8 * khalf;
        C[(size_t)(i0 + r) * HID + col] = acc[v] + bs;
    }
}

// ---------------------------------------------------------------------------
__global__ void colstats_kernel(const float* __restrict__ X,
                                const float* __restrict__ g, const float* __restrict__ b,
                                float* __restrict__ scale, float* __restrict__ shift) {
    __shared__ float ssum[256], ssq[256];
    const int c = blockIdx.x;
    float s = 0.f, q = 0.f;
    for (int r = threadIdx.x; r < NN; r += 256) {
        float v = X[(size_t)r * HID + c];
        s += v;
        q += v * v;
    }
    ssum[threadIdx.x] = s;
    ssq[threadIdx.x]  = q;
    __syncthreads();
    for (int off = 128; off; off >>= 1) {
        if (threadIdx.x < off) {
            ssum[threadIdx.x] += ssum[threadIdx.x + off];
            ssq[threadIdx.x]  += ssq[threadIdx.x + off];
        }
        __syncthreads();
    }
    if (threadIdx.x == 0) {
        float mean = ssum[0] * (1.f / NN);
        float var  = ssq[0] * (1.f / NN) - mean * mean;
        float gs   = g[c] * rsqrtf(var + BN_EPS);
        scale[c] = gs;
        shift[c] = b[c] - mean * gs;
    }
}

__global__ void bn_relu_kernel(const float* __restrict__ X,
                               const float* __restrict__ scale,
                               const float* __restrict__ shift,
                               float* __restrict__ H, float* __restrict__ H2) {
    int idx = blockIdx.x * blockDim.x + threadIdx.x;
    if (idx >= NN * HID) return;
    int c = idx & (HID - 1);
    float v = X[idx] * scale[c] + shift[c];
    v = (v > 0.f) ? v : 0.f;
    H[idx] = v;
    if (H2) H2[idx] = v;
}

__global__ void pool_kernel(const float* __restrict__ gp,
                            const float* __restrict__ h, float* __restrict__ out) {
    const int g = blockIdx.x, c = threadIdx.x;   // 8 x 128
    const float* gpr = gp + (size_t)g * NN;
    float acc = 0.f;
    for (int n = 0; n < NN; ++n) acc += gpr[n] * h[(size_t)n * HID + c];
    out[g * HID + c] = acc;
}

// ---------------------------------------------------------------------------
extern "C" void kernel_launch(void* const* d_in, const int* in_sizes, int n_in,
                              void* d_out, int out_size, void* d_ws, size_t ws_size,
                              hipStream_t stream) {
    // pytree leaf order: x, graph_pool, padded_nei, adj, eps,
    //   per layer: W_att, a_att, bn_b, bn_g, mlp_W1, mlp_W2, mlp_b1, mlp_b2, mlp_bn_b, mlp_bn_g
    const float* x   = (const float*)d_in[0];
    const float* gp  = (const float*)d_in[1];
    const float* adj = (const float*)d_in[3];
    const float* eps = (const float*)d_in[4];
    auto L = [&](int l, int k) -> const float* { return (const float*)d_in[5 + l * 10 + k]; };

    float* out = (float*)d_out;                 // [1024 pooled_h][4096*128 h_nodes]
    float* ws  = (float*)d_ws;
    float* s1    = ws;              ws += NN;
    float* s2    = ws;              ws += NN;
    float* maxv  = ws;              ws += 64;
    float* wa1   = ws;              ws += HID;
    float* wa2   = ws;              ws += HID;
    float* scale = ws;              ws += HID;
    float* shift = ws;              ws += HID;
    float* bufA  = ws;              ws += (size_t)NN * HID;  // pooled / t2
    float* bufB  = ws;              ws += (size_t)NN * HID;  // t1
    float* hbuf  = ws;              ws += (size_t)NN * HID;  // layer output

    const float* hin = x;
    for (int l = 0; l < NLAY; ++l) {
        prep_att_kernel<<<1, HID, 0, stream>>>(L(l, 0), L(l, 1), wa1, wa2);
        scores_kernel<<<NN / 8, 256, 0, stream>>>(hin, wa1, wa2, s1, s2);
        max_kernel<<<1, 256, 0, stream>>>(s2, maxv);
        agg_wmma_kernel<<<dim3(NN / 64, HID / 64), 128, 0, stream>>>(
            adj, hin, s1, s2, maxv, eps, l, bufA);
        gemm_wmma_kernel<0><<<dim3(NN / 16, HID / 16), 32, 0, stream>>>(
            bufA, L(l, 4), L(l, 6), nullptr, nullptr, bufB);
        colstats_kernel<<<HID, 256, 0, stream>>>(bufB, L(l, 9), L(l, 8), scale, shift);
        gemm_wmma_kernel<1><<<dim3(NN / 16, HID / 16), 32, 0, stream>>>(
            bufB, L(l, 5), L(l, 7), scale, shift, bufA);
        colstats_kernel<<<HID, 256, 0, stream>>>(bufA, L(l, 3), L(l, 2), scale, shift);
        bn_relu_kernel<<<(NN * HID + 255) / 256, 256, 0, stream>>>(
            bufA, scale, shift, hbuf, (l == NLAY - 1) ? (out + NG * HID) : nullptr);
        hin = hbuf;
    }
    pool_kernel<<<NG, HID, 0, stream>>>(gp, hbuf, out);
}